// ProtoNet_87634512708165
// MI455X (gfx1250) — compile-verified
//
#include <hip/hip_runtime.h>
#include <hip/hip_bf16.h>

// ---------------------------------------------------------------------------
// ProtoNet + soft-DTW loss for MI455X (gfx1250).
// fp32 throughout (matches reference); GEMM via V_WMMA_F32_16X16X4_F32.
// Problem constants from the reference harness.
// ---------------------------------------------------------------------------

typedef __attribute__((ext_vector_type(2))) float v2f;
typedef __attribute__((ext_vector_type(8))) float v8f;

#define CC      256
#define TT      8
#define VV      25
#define CTV     51200            // C*T*V
#define FF      6400             // V*C
#define NCLASS  20
#define NSUP    5
#define NQPC    15               // queries per class
#define SPC     20               // samples per class
#define NQ      300              // total queries
#define ROWS    2400             // NQ * T
#define COLS    160              // NCLASS * T
#define NPANEL  150              // ROWS/16
#define NCTILE  10               // COLS/16
#define KG      1600             // FF/4 : k-groups per WMMA chain
#define NB_DTW  6320             // 6000 cross + 300 xx + 20 yy
#define GAMMA_F 0.1f
#define BIGINF  1e10f

__device__ __forceinline__ int qglob(int n) {
    // query n (0..299) -> global sample index: class*20 + 5 + within-class
    return (n / NQPC) * SPC + NSUP + (n % NQPC);
}

// ---------------------------------------------------------------------------
// 1) proto[cls][c][t][v] = mean over 5 support samples
// ---------------------------------------------------------------------------
__global__ void proto_kernel(const float* __restrict__ in, float* __restrict__ proto) {
    int idx = blockIdx.x * blockDim.x + threadIdx.x;          // 1,024,000 total
    int cls = idx / CTV;
    int rem = idx - cls * CTV;
    const float* base = in + (size_t)cls * SPC * CTV + rem;
    float s = 0.f;
#pragma unroll
    for (int k = 0; k < NSUP; ++k) s += base[(size_t)k * CTV];
    proto[idx] = s * 0.2f;
}

// ---------------------------------------------------------------------------
// 2) Pack A (queries) into WMMA fragment order.
//    ISA 16x4 f32 A layout: lanes 0-15: M=lane, K = 4g,4g+1 (VGPR0,1)
//                           lanes16-31: M=lane-16, K = 4g+2,4g+3
//    Apack[p][g][lane] = (A[M][k0], A[M][k0+1])   -- coalesced b64 in GEMM
//    Feature mapping f = v*256 + c  ->  v = f>>8, c = f&255
// ---------------------------------------------------------------------------
__global__ void packA_kernel(const float* __restrict__ in, float* __restrict__ apackf) {
    int t = blockIdx.x * blockDim.x + threadIdx.x;            // 150*1600*32
    if (t >= NPANEL * KG * 32) return;
    int lane = t & 31;
    int g    = (t >> 5) % KG;
    int p    = t / (KG * 32);
    int Mrow = lane & 15;
    int kb   = g * 4 + ((lane >> 4) << 1);
    int row  = p * 16 + Mrow;                                  // 0..2399
    int n    = row >> 3;
    int i    = row & 7;
    const float* src = in + (size_t)qglob(n) * CTV + i * VV;
    v2f val;
    { int k = kb;     int v = k >> 8, c = k & 255; val.x = src[c * (TT*VV) + v]; }
    { int k = kb + 1; int v = k >> 8, c = k & 255; val.y = src[c * (TT*VV) + v]; }
    ((v2f*)apackf)[t] = val;
}

// ---------------------------------------------------------------------------
// 3) Pack B (protos) into WMMA fragment order (B = Yp^T, 4x16 per step).
//    Assumed B layout mirrors C/D striping: lanes 0-15: N=lane, K=4g,4g+1;
//    lanes 16-31: N=lane-16, K=4g+2,4g+3.
// ---------------------------------------------------------------------------
__global__ void packB_kernel(const float* __restrict__ proto, float* __restrict__ bpackf) {
    int t = blockIdx.x * blockDim.x + threadIdx.x;            // 10*1600*32
    if (t >= NCTILE * KG * 32) return;
    int lane = t & 31;
    int g    = (t >> 5) % KG;
    int ct   = t / (KG * 32);
    int N    = lane & 15;
    int kb   = g * 4 + ((lane >> 4) << 1);
    int col  = ct * 16 + N;                                    // 0..159
    int mcls = col >> 3;
    int j    = col & 7;
    const float* src = proto + (size_t)mcls * CTV + j * VV;
    v2f val;
    { int k = kb;     int v = k >> 8, c = k & 255; val.x = src[c * (TT*VV) + v]; }
    { int k = kb + 1; int v = k >> 8, c = k & 255; val.y = src[c * (TT*VV) + v]; }
    ((v2f*)bpackf)[t] = val;
}

// ---------------------------------------------------------------------------
// 4) Self-grams: Gxx[n][i][j] (300x8x8) and Gyy[m][i][j] (20x8x8).
//    One wave per (entity, i, j); wave32 butterfly reduction.
//    Squared norms are the diagonals -> no separate norm pass needed.
// ---------------------------------------------------------------------------
__global__ void selfgram_kernel(const float* __restrict__ in, const float* __restrict__ proto,
                                float* __restrict__ gxx, float* __restrict__ gyy) {
    int wid  = (blockIdx.x * blockDim.x + threadIdx.x) >> 5;   // 320*64 waves
    int lane = threadIdx.x & 31;
    if (wid >= (NQ + NCLASS) * 64) return;
    int e  = wid >> 6;
    int ij = wid & 63;
    int i  = ij >> 3;
    int j  = ij & 7;
    const float* base;
    float* dst;
    if (e < NQ) { base = in + (size_t)qglob(e) * CTV;       dst = gxx + e * 64 + ij; }
    else        { base = proto + (size_t)(e - NQ) * CTV;    dst = gyy + (e - NQ) * 64 + ij; }
    const float* pi = base + i * VV;
    const float* pj = base + j * VV;
    float s = 0.f;
    for (int f = lane; f < FF; f += 32) {
        int v = f >> 8, c = f & 255;
        int off = c * (TT*VV) + v;
        s += pi[off] * pj[off];
    }
#pragma unroll
    for (int off = 16; off > 0; off >>= 1) s += __shfl_xor(s, off, 32);
    if (lane == 0) *dst = s;
}

// ---------------------------------------------------------------------------
// 5) Cross gram Gxy (2400x160) = Apack x Bpack^T via V_WMMA_F32_16X16X4_F32.
//    One wave per 16x16 tile, 1500 waves, K-chain of 1600 WMMAs unrolled x4.
// ---------------------------------------------------------------------------
__global__ void __launch_bounds__(128) gemm_wmma_kernel(const float* __restrict__ apackf,
                                                        const float* __restrict__ bpackf,
                                                        float* __restrict__ gxy) {
    int wave = threadIdx.x >> 5;
    int lane = threadIdx.x & 31;
    int tile = blockIdx.x * 4 + wave;                          // 0..1499 (exact)
    int p    = tile / NCTILE;
    int ct   = tile % NCTILE;
    const v2f* Ap = ((const v2f*)apackf) + (size_t)p  * KG * 32;
    const v2f* Bp = ((const v2f*)bpackf) + (size_t)ct * KG * 32;

    v8f acc = {};
    for (int g = 0; g < KG; g += 4) {
        v2f a0 = Ap[(g + 0) * 32 + lane];
        v2f b0 = Bp[(g + 0) * 32 + lane];
        v2f a1 = Ap[(g + 1) * 32 + lane];
        v2f b1 = Bp[(g + 1) * 32 + lane];
        v2f a2 = Ap[(g + 2) * 32 + lane];
        v2f b2 = Bp[(g + 2) * 32 + lane];
        v2f a3 = Ap[(g + 3) * 32 + lane];
        v2f b3 = Bp[(g + 3) * 32 + lane];
        __builtin_prefetch(&Ap[(g + 32) * 32 + lane], 0, 0);   // -> global_prefetch_b8
        __builtin_prefetch(&Bp[(g + 32) * 32 + lane], 0, 0);
        acc = __builtin_amdgcn_wmma_f32_16x16x4_f32(false, a0, false, b0, (short)0, acc, false, false);
        acc = __builtin_amdgcn_wmma_f32_16x16x4_f32(false, a1, false, b1, (short)0, acc, false, false);
        acc = __builtin_amdgcn_wmma_f32_16x16x4_f32(false, a2, false, b2, (short)0, acc, false, false);
        acc = __builtin_amdgcn_wmma_f32_16x16x4_f32(false, a3, false, b3, (short)0, acc, false, false);
    }
    // D layout: VGPR r -> (M = r | r+8) by lane half, N = lane&15
    int rbase = p * 16 + ((lane >> 4) << 3);
    int col   = ct * 16 + (lane & 15);
#pragma unroll
    for (int r = 0; r < 8; ++r)
        gxy[(size_t)(rbase + r) * COLS + col] = acc[r];
}

// ---------------------------------------------------------------------------
// 6) soft-DTW over 6320 8x8 cost matrices, built on the fly from grams.
//    b < 6000 : Dxy (query n, proto m);  6000..6299 : Dxx;  6300.. : Dyy
// ---------------------------------------------------------------------------
__global__ void dtw_kernel(const float* __restrict__ gxy, const float* __restrict__ gxx,
                           const float* __restrict__ gyy, float* __restrict__ rout) {
    int b = blockIdx.x * blockDim.x + threadIdx.x;
    if (b >= NB_DTW) return;

    const float* gp; int base, rstride;
    float xd[TT], yd[TT];
    if (b < 6000) {
        int n = b / NCLASS, m = b % NCLASS;
#pragma unroll
        for (int t = 0; t < TT; ++t) { xd[t] = gxx[n * 64 + t * 9]; yd[t] = gyy[m * 64 + t * 9]; }
        gp = gxy; base = (n * TT) * COLS + m * TT; rstride = COLS;
    } else if (b < 6000 + NQ) {
        int n = b - 6000;
#pragma unroll
        for (int t = 0; t < TT; ++t) { xd[t] = gxx[n * 64 + t * 9]; yd[t] = xd[t]; }
        gp = gxx; base = n * 64; rstride = TT;
    } else {
        int m = b - 6000 - NQ;
#pragma unroll
        for (int t = 0; t < TT; ++t) { xd[t] = gyy[m * 64 + t * 9]; yd[t] = xd[t]; }
        gp = gyy; base = m * 64; rstride = TT;
    }

    float Rprev[TT + 1], Rcur[TT + 1];
    Rprev[0] = 0.f;
#pragma unroll
    for (int j = 1; j <= TT; ++j) Rprev[j] = BIGINF;

    for (int i = 1; i <= TT; ++i) {
        Rcur[0] = BIGINF;
#pragma unroll
        for (int j = 1; j <= TT; ++j) {
            float d  = xd[i - 1] + yd[j - 1] - 2.f * gp[base + (i - 1) * rstride + (j - 1)];
            float a  = Rprev[j];
            float bb = Rcur[j - 1];
            float c  = Rprev[j - 1];
            float mn = fminf(a, fminf(bb, c));
            float s  = __expf((mn - a) / GAMMA_F) + __expf((mn - bb) / GAMMA_F)
                     + __expf((mn - c) / GAMMA_F);
            Rcur[j] = d + mn - GAMMA_F * __logf(s);
        }
#pragma unroll
        for (int j = 0; j <= TT; ++j) Rprev[j] = Rcur[j];
    }
    rout[b] = Rprev[TT];
}

// ---------------------------------------------------------------------------
// 7) dist -> log_softmax -> mean NLL; single block.
// ---------------------------------------------------------------------------
__global__ void loss_kernel(const float* __restrict__ rall, float* __restrict__ out) {
    __shared__ float sdata[512];
    int tid = threadIdx.x;
    float contrib = 0.f;
    if (tid < NQ) {
        int q = tid, cls = q / NQPC;
        float rxx = rall[6000 + q];
        float dist[NCLASS];
        float mx = -3.402823466e38f;
#pragma unroll
        for (int m = 0; m < NCLASS; ++m) {
            dist[m] = rall[q * NCLASS + m] - 0.5f * (rxx + rall[6300 + m]);
            mx = fmaxf(mx, -dist[m]);
        }
        float s = 0.f;
#pragma unroll
        for (int m = 0; m < NCLASS; ++m) s += __expf(-dist[m] - mx);
        float lse  = mx + __logf(s);
        float logp = (-dist[cls]) - lse;
        contrib = -logp;
    }
    sdata[tid] = contrib;
    __syncthreads();
#pragma unroll
    for (int off = 256; off > 0; off >>= 1) {
        if (tid < off) sdata[tid] += sdata[tid + off];
        __syncthreads();
    }
    if (tid == 0) out[0] = sdata[0] / (float)NQ;
}

// ---------------------------------------------------------------------------
// Launch. Workspace layout (floats):
//   proto 1,024,000 | Apack 15,360,000 | Bpack 1,024,000 | Gxy 384,000
//   Gxx 19,200 | Gyy 1,280 | rall 6,320   => ~71.3 MB total
// ---------------------------------------------------------------------------
extern "C" void kernel_launch(void* const* d_in, const int* in_sizes, int n_in,
                              void* d_out, int out_size, void* d_ws, size_t ws_size,
                              hipStream_t stream) {
    (void)in_sizes; (void)n_in; (void)out_size; (void)ws_size;
    const float* in = (const float*)d_in[0];
    float* ws    = (float*)d_ws;
    float* proto = ws;
    float* apack = proto + 1024000;
    float* bpack = apack + 15360000;
    float* gxy   = bpack + 1024000;
    float* gxx   = gxy   + (size_t)ROWS * COLS;   // 384,000
    float* gyy   = gxx   + NQ * 64;               // 19,200
    float* rall  = gyy   + NCLASS * 64;           // 1,280
    float* out   = (float*)d_out;

    proto_kernel   <<<4000,  256, 0, stream>>>(in, proto);
    packA_kernel   <<<30000, 256, 0, stream>>>(in, apack);
    packB_kernel   <<<2000,  256, 0, stream>>>(proto, bpack);
    selfgram_kernel<<<2560,  256, 0, stream>>>(in, proto, gxx, gyy);
    gemm_wmma_kernel<<<375,  128, 0, stream>>>(apack, bpack, gxy);
    dtw_kernel     <<<25,    256, 0, stream>>>(gxy, gxx, gyy, rall);
    loss_kernel    <<<1,     512, 0, stream>>>(rall, out);
}